// RNN_42820823941693
// MI455X (gfx1250) — compile-verified
//
#include <hip/hip_runtime.h>
#include <stdint.h>

// Problem constants (fixed by the reference).
#define T_LEN 2048
#define B_SZ  8192
// Pipeline config: 16 timesteps per chunk, triple-buffered LDS, 32-thread
// (single wave32) blocks, 2 batch elements per thread.
#define TS    16
#define NBUF  3
#define BLK   32
#define EPT   2
#define NC    (T_LEN / TS)

// Real clang vector types (HIP float2/float4 are structs and are rejected by
// __builtin_nontemporal_store).
typedef float v2f __attribute__((ext_vector_type(2)));
typedef float v4f __attribute__((ext_vector_type(4)));

// ---------------- CDNA5-specific primitives ----------------

__device__ __forceinline__ float fast_tanh(float x) {
#if __has_builtin(__builtin_amdgcn_tanhf)
    return __builtin_amdgcn_tanhf(x);
#elif __has_builtin(__builtin_amdgcn_tanh_f32)
    return __builtin_amdgcn_tanh_f32(x);
#else
    float r;
    asm("v_tanh_f32 %0, %1" : "=v"(r) : "v"(x));
    return r;
#endif
}

// Wait until ASYNCcnt <= n (async LDS-loads complete in order).
#define WAIT_ASYNC(n) asm volatile("s_wait_asynccnt %0" :: "n"(n) : "memory")
// Drain DS reads before re-targeting an LDS buffer with async writes.
#define WAIT_DS0()    asm volatile("s_wait_dscnt 0x0" ::: "memory")

// Async DMA: 16 bytes per lane, global -> LDS, tracked by ASYNCcnt.
__device__ __forceinline__ void async_ld_b128(const void* g, uint32_t lds_byte) {
    asm volatile("global_load_async_to_lds_b128 %0, %1, off"
                 :: "v"(lds_byte), "v"(g) : "memory");
}

// Issue one chunk: TS rows; each lane fetches its own 16B (x for its 2 batch
// elements at one timestep). Row stride in global memory is B*I*4 = B*8 bytes.
__device__ __forceinline__ void issue_chunk(const char* gl, uint32_t lds) {
#pragma unroll
    for (int r = 0; r < TS; ++r)
        async_ld_b128(gl + (size_t)r * (B_SZ * 8), lds + (uint32_t)r * (BLK * 16));
}

// ---------------- GRU step (one chain of H=2) ----------------
// r/z rows of Wi/Wh and merged bias Bc are pre-scaled by 0.5 so that
// sigmoid(v) = 0.5 + 0.5*tanh(0.5*v) costs a single v_tanh_f32 each.
// fw*/fb are pre-scaled by log2(e) so the output is one v_exp_f32.
__device__ __forceinline__ float gru_step(
    float& h0, float& h1, float x0, float x1,
    const float (&Wi)[6][2], const float (&Wh)[6][2],
    const float (&Bc)[4], const float (&Bin)[2], const float (&Bhn)[2],
    float fw0, float fw1, float fb)
{
    float ar0 = fmaf(Wi[0][1], x1, fmaf(Wi[0][0], x0, Bc[0]));
    ar0       = fmaf(Wh[0][1], h1, fmaf(Wh[0][0], h0, ar0));
    float ar1 = fmaf(Wi[1][1], x1, fmaf(Wi[1][0], x0, Bc[1]));
    ar1       = fmaf(Wh[1][1], h1, fmaf(Wh[1][0], h0, ar1));
    float az0 = fmaf(Wi[2][1], x1, fmaf(Wi[2][0], x0, Bc[2]));
    az0       = fmaf(Wh[2][1], h1, fmaf(Wh[2][0], h0, az0));
    float az1 = fmaf(Wi[3][1], x1, fmaf(Wi[3][0], x0, Bc[3]));
    az1       = fmaf(Wh[3][1], h1, fmaf(Wh[3][0], h0, az1));

    float r0 = fmaf(0.5f, fast_tanh(ar0), 0.5f);
    float r1 = fmaf(0.5f, fast_tanh(ar1), 0.5f);
    float z0 = fmaf(0.5f, fast_tanh(az0), 0.5f);
    float z1 = fmaf(0.5f, fast_tanh(az1), 0.5f);

    float in0 = fmaf(Wi[4][1], x1, fmaf(Wi[4][0], x0, Bin[0]));
    float in1 = fmaf(Wi[5][1], x1, fmaf(Wi[5][0], x0, Bin[1]));
    float hn0 = fmaf(Wh[4][1], h1, fmaf(Wh[4][0], h0, Bhn[0]));
    float hn1 = fmaf(Wh[5][1], h1, fmaf(Wh[5][0], h0, Bhn[1]));

    float n0 = fast_tanh(fmaf(r0, hn0, in0));
    float n1 = fast_tanh(fmaf(r1, hn1, in1));

    h0 = fmaf(z0, h0 - n0, n0);   // (1-z)*n + z*h
    h1 = fmaf(z1, h1 - n1, n1);

    return __builtin_amdgcn_exp2f(fmaf(fw1, h1, fmaf(fw0, h0, fb)));
}

// ---------------- Kernel ----------------

__global__ __launch_bounds__(BLK) void
gru_scan_kernel(const float* __restrict__ x,
                const float* __restrict__ w_ih, const float* __restrict__ w_hh,
                const float* __restrict__ b_ih, const float* __restrict__ b_hh,
                const float* __restrict__ fc_w, const float* __restrict__ fc_b,
                float* __restrict__ out)
{
    __shared__ v4f sx[NBUF][TS][BLK];

    const int tid = threadIdx.x;
    const int b   = blockIdx.x * (BLK * EPT) + tid * EPT; // first of 2 batch elems

    // ---- uniform weight prep (scalarized by the compiler) ----
    float Wi[6][2], Wh[6][2], Bc[4], Bin[2], Bhn[2];
#pragma unroll
    for (int k = 0; k < 4; ++k) {   // r, z gate rows: pre-scale by 0.5
        Wi[k][0] = 0.5f * w_ih[2 * k + 0];
        Wi[k][1] = 0.5f * w_ih[2 * k + 1];
        Wh[k][0] = 0.5f * w_hh[2 * k + 0];
        Wh[k][1] = 0.5f * w_hh[2 * k + 1];
        Bc[k]    = 0.5f * (b_ih[k] + b_hh[k]);
    }
#pragma unroll
    for (int k = 4; k < 6; ++k) {   // n gate rows: biases kept separate (r * h_n)
        Wi[k][0] = w_ih[2 * k + 0];
        Wi[k][1] = w_ih[2 * k + 1];
        Wh[k][0] = w_hh[2 * k + 0];
        Wh[k][1] = w_hh[2 * k + 1];
        Bin[k - 4] = b_ih[k];
        Bhn[k - 4] = b_hh[k];
    }
    const float LOG2E = 1.44269504088896340736f;
    const float fw0 = fc_w[0] * LOG2E;
    const float fw1 = fc_w[1] * LOG2E;
    const float fb  = fc_b[0] * LOG2E;

    // ---- async streaming setup ----
    // Lane's global base: &x[t=0][b][0]  (8 bytes per batch elem, 16B per lane)
    const char*  gl      = (const char*)x + (size_t)b * 8;
    const size_t CSTRIDE = (size_t)TS * B_SZ * 8;                    // bytes per chunk
    const uint32_t lds0  = (uint32_t)(uintptr_t)&sx[0][0][tid];      // LDS byte addr
    const uint32_t BUFB  = TS * BLK * 16;                            // bytes per buffer

    issue_chunk(gl, lds0);                                           // chunk 0
    issue_chunk(gl + CSTRIDE, lds0 + BUFB);                          // chunk 1

    float h0a = 0.f, h1a = 0.f, h0b = 0.f, h1b = 0.f;

#pragma unroll 1
    for (int c = 0; c < NC; ++c) {
        if (c + 2 < NC) {
            WAIT_DS0();  // prior reads of this buffer fully retired
            issue_chunk(gl + (size_t)(c + 2) * CSTRIDE,
                        lds0 + (uint32_t)((c + 2) % NBUF) * BUFB);
            WAIT_ASYNC(2 * TS);   // chunk c landed (in-order completion)
        } else if (c + 2 == NC) {
            WAIT_ASYNC(TS);
        } else {
            WAIT_ASYNC(0);
        }

        const v4f* buf  = &sx[c % NBUF][0][tid];
        float*     orow = out + (size_t)c * TS * B_SZ + b;

#pragma unroll
        for (int r = 0; r < TS; ++r) {
            v4f xv = buf[r * BLK];   // ds_load_b128: both elems' x at step t
            float oA = gru_step(h0a, h1a, xv.x, xv.y, Wi, Wh, Bc, Bin, Bhn, fw0, fw1, fb);
            float oB = gru_step(h0b, h1b, xv.z, xv.w, Wi, Wh, Bc, Bin, Bhn, fw0, fw1, fb);
            v2f o2 = {oA, oB};
            __builtin_nontemporal_store(o2, (v2f*)(orow + (size_t)r * B_SZ));
        }
    }

    // h_last: [1, B, H] appended after out[T, B, 1]; 2 elems * H=2 = 4 floats
    v4f hl = {h0a, h1a, h0b, h1b};
    __builtin_nontemporal_store(hl, (v4f*)(out + (size_t)T_LEN * B_SZ + (size_t)b * 2));
}

// ---------------- Host launch ----------------

extern "C" void kernel_launch(void* const* d_in, const int* in_sizes, int n_in,
                              void* d_out, int out_size, void* d_ws, size_t ws_size,
                              hipStream_t stream) {
    (void)in_sizes; (void)n_in; (void)out_size; (void)d_ws; (void)ws_size;
    const float* x    = (const float*)d_in[0];
    const float* w_ih = (const float*)d_in[1];
    const float* w_hh = (const float*)d_in[2];
    const float* b_ih = (const float*)d_in[3];
    const float* b_hh = (const float*)d_in[4];
    const float* fc_w = (const float*)d_in[5];
    const float* fc_b = (const float*)d_in[6];
    float* out = (float*)d_out;

    dim3 grid(B_SZ / (BLK * EPT));   // 128 single-wave blocks
    dim3 block(BLK);                 // one wave32 per workgroup: no barriers needed
    gru_scan_kernel<<<grid, block, 0, stream>>>(x, w_ih, w_hh, b_ih, b_hh, fc_w, fc_b, out);
}